// LIAFSpike_67242007986186
// MI455X (gfx1250) — compile-verified
//
#include <hip/hip_runtime.h>

// LIAF spiking-neuron scan, T=4, tau=0.5, thresh=1.0.
// Pure HBM-streaming op: 268 MB total traffic -> ~11.5 us floor @ 23.3 TB/s.
// Strategy: one float4 column per lane, all 4 time steps staged via the CDNA5
// async global->LDS data mover (ASYNCcnt), scan in registers, NT b128 stores.

#define T_STEPS 4
#define TAU     0.5f
#define THRESH  1.0f

typedef float v4f __attribute__((ext_vector_type(4)));

__device__ __forceinline__ float liaf_step(float& mem, float x) {
    float m   = __builtin_fmaf(mem, TAU, x);   // leaky integrate
    float out = fmaxf(m - THRESH, 0.0f);       // LIAF analog output
    mem       = (m >= THRESH) ? 0.0f : m;      // hard reset on spike
    return out;
}

__global__ __launch_bounds__(256) void liaf_scan_kernel(const float* __restrict__ xp,
                                                        float* __restrict__ op,
                                                        long long n4) {
    // 4 KB per time step, 16 KB total LDS per block (of 320 KB/WGP -> no
    // occupancy pressure).
    __shared__ __align__(16) v4f tile[T_STEPS][256];

    const long long i = (long long)blockIdx.x * 256 + threadIdx.x; // float4 column id
    if (i >= n4) return;

    const v4f* __restrict__ xv = (const v4f*)xp;
    v4f* __restrict__ ov = (v4f*)op;

    // Stage this lane's 4 time-step rows into LDS with the gfx1250 async
    // global->LDS path. Per-lane LDS destination = low 32 bits of the generic
    // pointer into the shared array (DS-space offset).
#pragma unroll
    for (int t = 0; t < T_STEPS; ++t) {
        unsigned loff = (unsigned)(unsigned long long)(void*)&tile[t][threadIdx.x];
        unsigned long long ga = (unsigned long long)(xv + (long long)t * n4 + i);
        asm volatile("global_load_async_to_lds_b128 %0, %1, off"
                     :: "v"(loff), "v"(ga)
                     : "memory");
    }
    // Drain the async counter: all 4 tiles resident in LDS.
    asm volatile("s_wait_asynccnt 0" ::: "memory");

    v4f xs0 = tile[0][threadIdx.x];
    v4f xs1 = tile[1][threadIdx.x];
    v4f xs2 = tile[2][threadIdx.x];
    v4f xs3 = tile[3][threadIdx.x];

    v4f o0, o1, o2, o3;
#pragma unroll
    for (int c = 0; c < 4; ++c) {
        float mem = 0.0f;
        o0[c] = liaf_step(mem, xs0[c]);
        o1[c] = liaf_step(mem, xs1[c]);
        o2[c] = liaf_step(mem, xs2[c]);
        o3[c] = liaf_step(mem, xs3[c]);
    }

    // Streaming output, no reuse, working set > 192 MB L2 -> non-temporal.
    __builtin_nontemporal_store(o0, ov + i);
    __builtin_nontemporal_store(o1, ov + n4 + i);
    __builtin_nontemporal_store(o2, ov + 2LL * n4 + i);
    __builtin_nontemporal_store(o3, ov + 3LL * n4 + i);
}

extern "C" void kernel_launch(void* const* d_in, const int* in_sizes, int n_in,
                              void* d_out, int out_size, void* d_ws, size_t ws_size,
                              hipStream_t stream) {
    (void)n_in; (void)out_size; (void)d_ws; (void)ws_size;

    const float* x = (const float*)d_in[0];
    float* out = (float*)d_out;

    const long long total = (long long)in_sizes[0];     // T*B*C*H*W = 33,554,432
    const long long n4    = total / (T_STEPS * 4);      // float4 columns per step = 2^21

    dim3 block(256);
    dim3 grid((unsigned)((n4 + 255) / 256));            // 8192 blocks, exact fit
    liaf_scan_kernel<<<grid, block, 0, stream>>>(x, out, n4);
}